// MinkowskiGlobalPooling_42949672960750
// MI455X (gfx1250) — compile-verified
//
#include <hip/hip_runtime.h>
#include <hip/hip_bf16.h>
#include <stdint.h>

// MinkowskiGlobalPooling (average=True) for MI455X / gfx1250.
// Segment-sum as one-hot GEMM on V_WMMA_F32_16X16X4_F32.
// - feats streamed once with non-temporal loads (1 GB >> 192 MB L2)
// - batch indices (wave-uniform per quad) read via SMEM scalar loads
// - sorted-index 16-wide batch window, rare uniform rebase + ds_add_f32 flush
// - deterministic two-stage reduction via per-block partials in d_ws
// - segment counts via binary search on the sorted index (no extra 32 MB scan)

typedef __attribute__((ext_vector_type(2))) float v2f;
typedef __attribute__((ext_vector_type(8))) float v8f;

#define CHN 64           // channels (reference: 64)
#define MAXB 32          // batches (reference: 32)
#define LDS_ROWS 48      // base(<=31) + 15 window rows
#define POOL_ELEMS (MAXB * CHN)

__device__ __forceinline__ void flush_acc(float* lds, v8f& acc, int base, int hi, int ch)
{
    #pragma unroll
    for (int r = 0; r < 8; ++r)
        atomicAdd(&lds[(base + r + (hi << 3)) * CHN + ch], acc[r]);
    acc = (v8f){0.f, 0.f, 0.f, 0.f, 0.f, 0.f, 0.f, 0.f};
}

__global__ __launch_bounds__(256) void pool_partial_kernel(
    const float* __restrict__ feats,
    const long long* __restrict__ batch,   // int64, sorted ascending
    float* __restrict__ partial,           // [nblocks][MAXB*CHN]
    int npoints, int nblocks)
{
    __shared__ float ldsPool[LDS_ROWS * CHN];

    const int tid = threadIdx.x;
    for (int i = tid; i < LDS_ROWS * CHN; i += 256) ldsPool[i] = 0.0f;
    __syncthreads();

    const int lane  = tid & 31;
    const int wave  = __builtin_amdgcn_readfirstlane(tid >> 5); // scalar wave id
    const int grp   = wave >> 2;          // 0..1 : point group
    const int cbase = (wave & 3) << 4;    // 0,16,32,48 : channel group
    const int m     = lane & 15;          // batch-row slot / channel-in-group
    const int hi    = lane >> 4;          // half-wave selector (K split)
    const int ch    = cbase + m;          // this lane's channel

    // Per-block contiguous point range, split into 2 point groups.
    const int chunk  = (npoints + nblocks - 1) / nblocks;
    const int bstart = blockIdx.x * chunk;
    const int bend   = min(bstart + chunk, npoints);
    const int half   = (chunk + 1) >> 1;
    const int gstart = min(bstart + grp * half, bend);
    const int gend   = min(gstart + half, bend);

    v8f acc  = {0.f, 0.f, 0.f, 0.f, 0.f, 0.f, 0.f, 0.f};
    int base = 0;  // batch-window base (always <= every remaining batch id)

    // Uniform streaming pointers -> SMEM index loads + GVS feature loads
    const long long* bp = batch + gstart;
    const float*     fp = feats + (size_t)gstart * CHN;
    const int voff = (hi << 1) * CHN + ch;   // lane-constant element offset (k0 of quad)

    const int count = gend - gstart;
    const int nfull = count & ~7;

    for (int it = 0; it < nfull; it += 8) {
        // Wave-uniform batch ids for 8 points (scalar loads).
        const int sb0 = (int)bp[0], sb1 = (int)bp[1], sb2 = (int)bp[2], sb3 = (int)bp[3];
        const int sb4 = (int)bp[4], sb5 = (int)bp[5], sb6 = (int)bp[6], sb7 = (int)bp[7];

        // Non-temporal streaming feature loads (constant ioffsets, issue early).
        const float f0 = __builtin_nontemporal_load(fp + voff);
        const float f1 = __builtin_nontemporal_load(fp + voff + CHN);
        const float f2 = __builtin_nontemporal_load(fp + voff + 4 * CHN);
        const float f3 = __builtin_nontemporal_load(fp + voff + 5 * CHN);

        // Rare, wave-uniform window rebase (sorted index).
        if (sb7 - base >= 16) {
            flush_acc(ldsPool, acc, base, hi, ch);
            base = sb0;
            if (sb7 - base >= 16) {
                // Pathological: 8 points span >16 batches. Points above the
                // window get exact scalar adds here (their a-values are 0).
                const int t  = base + 16;
                const int b0 = hi ? sb2 : sb0, b1 = hi ? sb3 : sb1;
                const int b2 = hi ? sb6 : sb4, b3 = hi ? sb7 : sb5;
                if (b0 >= t) atomicAdd(&ldsPool[b0 * CHN + ch], f0);
                if (b1 >= t) atomicAdd(&ldsPool[b1 * CHN + ch], f1);
                if (b2 >= t) atomicAdd(&ldsPool[b2 * CHN + ch], f2);
                if (b3 >= t) atomicAdd(&ldsPool[b3 * CHN + ch], f3);
            }
        }

        // One-hot A from SGPR batch ids: row m fires iff b == base + m.
        const int tgt = base + m;
        {
            const int b0 = hi ? sb2 : sb0;
            const int b1 = hi ? sb3 : sb1;
            const v2f Am = { (b0 == tgt) ? 1.0f : 0.0f,
                             (b1 == tgt) ? 1.0f : 0.0f };
            const v2f Bm = { f0, f1 };
            acc = __builtin_amdgcn_wmma_f32_16x16x4_f32(
                      false, Am, false, Bm, (short)0, acc, false, false);
        }
        {
            const int b2 = hi ? sb6 : sb4;
            const int b3 = hi ? sb7 : sb5;
            const v2f Am = { (b2 == tgt) ? 1.0f : 0.0f,
                             (b3 == tgt) ? 1.0f : 0.0f };
            const v2f Bm = { f2, f3 };
            acc = __builtin_amdgcn_wmma_f32_16x16x4_f32(
                      false, Am, false, Bm, (short)0, acc, false, false);
        }

        bp += 8;
        fp += 8 * CHN;
    }

    // Guarded tail (< 8 points): clamped addresses, zeroed A for invalid lanes.
    for (int p = gstart + nfull; p < gend; p += 4) {
        const int pl    = min(p + 3, gend - 1);
        const int blast = (int)batch[pl];
        if (blast - base >= 16) {
            flush_acc(ldsPool, acc, base, hi, ch);
            base = (int)batch[p];
        }
        const int  k0  = hi << 1;
        const int  p0  = p + k0;
        const int  p1  = p0 + 1;
        const bool v0g = p0 < gend;
        const bool v1g = p1 < gend;
        const int  q0  = v0g ? p0 : gstart;
        const int  q1  = v1g ? p1 : gstart;

        const int bb0 = (int)batch[q0];
        const int bb1 = (int)batch[q1];
        const float f0 = __builtin_nontemporal_load(&feats[(size_t)q0 * CHN + ch]);
        const float f1 = __builtin_nontemporal_load(&feats[(size_t)q1 * CHN + ch]);

        const int  l0  = bb0 - base;
        const int  l1  = bb1 - base;
        const bool in0 = v0g && ((unsigned)l0 < 16u);
        const bool in1 = v1g && ((unsigned)l1 < 16u);

        if (v0g && !in0) atomicAdd(&ldsPool[bb0 * CHN + ch], f0);
        if (v1g && !in1) atomicAdd(&ldsPool[bb1 * CHN + ch], f1);

        const v2f Am = { (in0 && (l0 == m)) ? 1.0f : 0.0f,
                         (in1 && (l1 == m)) ? 1.0f : 0.0f };
        const v2f Bm = { f0, f1 };
        acc = __builtin_amdgcn_wmma_f32_16x16x4_f32(
                  false, Am, false, Bm, (short)0, acc, false, false);
    }

    // Final flush of the accumulator window.
    #pragma unroll
    for (int r = 0; r < 8; ++r)
        atomicAdd(&ldsPool[(base + r + (hi << 3)) * CHN + ch], acc[r]);
    __syncthreads();

    // Emit this block's partial pooled sums (plain coalesced stores).
    float* outp = partial + (size_t)blockIdx.x * POOL_ELEMS;
    for (int i = tid; i < POOL_ELEMS; i += 256)
        outp[i] = ldsPool[i];
}

__global__ __launch_bounds__(256) void pool_finalize_kernel(
    const float* __restrict__ partial,
    const long long* __restrict__ batch,
    float* __restrict__ out,
    int nblocks, int npoints, int nbatch)
{
    const int t = blockIdx.x * 256 + threadIdx.x;
    if (t >= nbatch * CHN) return;
    const int b = t / CHN;

    float s = 0.0f;
    for (int k = 0; k < nblocks; ++k)
        s += partial[(size_t)k * POOL_ELEMS + t];

    // Segment counts via binary search on the sorted int64 index.
    int lo0 = 0, hi0 = npoints;
    while (lo0 < hi0) {
        const int mid = (lo0 + hi0) >> 1;
        if (batch[mid] < (long long)b) lo0 = mid + 1; else hi0 = mid;
    }
    int lo1 = 0, hi1 = npoints;
    while (lo1 < hi1) {
        const int mid = (lo1 + hi1) >> 1;
        if (batch[mid] < (long long)(b + 1)) lo1 = mid + 1; else hi1 = mid;
    }
    const float cnt = (float)(lo1 - lo0);
    out[t] = s / fmaxf(cnt, 1.0f);
}

extern "C" void kernel_launch(void* const* d_in, const int* in_sizes, int n_in,
                              void* d_out, int out_size, void* d_ws, size_t ws_size,
                              hipStream_t stream)
{
    const float*     feats = (const float*)d_in[0];
    const long long* batch = (const long long*)d_in[1];  // jnp.int64
    // d_in[2] = num_batches scalar; nbatch derived from out_size instead.

    const int npoints = in_sizes[0] / CHN;
    const int nbatch  = out_size / CHN;       // 32 for the reference

    float* partial = (float*)d_ws;
    const size_t per = (size_t)POOL_ELEMS * sizeof(float);
    int nblocks = 1024;
    if ((size_t)nblocks * per > ws_size) {
        nblocks = (int)(ws_size / per);
        if (nblocks < 1) nblocks = 1;
    }

    hipLaunchKernelGGL(pool_partial_kernel, dim3(nblocks), dim3(256), 0, stream,
                       feats, batch, partial, npoints, nblocks);

    const int tot = nbatch * CHN;
    hipLaunchKernelGGL(pool_finalize_kernel, dim3((tot + 255) / 256), dim3(256), 0, stream,
                       partial, batch, (float*)d_out, nblocks, npoints, nbatch);
}